// S_ANODE_Linear_80109730005132
// MI455X (gfx1250) — compile-verified
//
#include <hip/hip_runtime.h>
#include <hip/hip_bf16.h>

typedef float v2f __attribute__((ext_vector_type(2)));
typedef float v8f __attribute__((ext_vector_type(8)));

#define DD   84      // ODE state dim (76 + 8 aug)
#define DH   85      // homogeneous (state + bias row)
#define NR   96      // padded tile rows (6 x 16)
#define PSTR 100     // precompute LDS stride (100%64=36, gcd4 -> 16 lanes, 16 banks)
#define WSTR 132     // W_pen^T LDS stride
#define FF   76      // real input features
#define HH   128     // penultimate width
#define CC   5       // classes
#define HSTEP 0.05f  // 1/ODE_STEPS

// Collapsed affine map, written by precompute, read by main kernel.
__device__ float g_E[FF * HH];   // E = (T+I)[0:76,:] @ W_pen^T   [76][128]
__device__ float g_bp[HH];       // b' = c @ W_pen^T + b_pen      [128]

// ---------------------------------------------------------------------------
// Kernel 1: collapse RK4(20 steps) of the linear ODE into an affine map and
// fold the skip-add + W_pen.  All matmuls run on the WMMA pipe:
//   K-chain (3 matmuls, K=84), Sh^20 by square-and-multiply (5 matmuls, K=88),
//   (T+I) @ W_pen^T (1 matmul, K=84).  Single workgroup, everything in LDS.
// ---------------------------------------------------------------------------
__global__ __launch_bounds__(256) void anode_precompute(
    const float* __restrict__ Wode, const float* __restrict__ bode,
    const float* __restrict__ Wpen, const float* __restrict__ bpen)
{
  __shared__ __align__(16) float B0[NR * PSTR];  // Aaug / P5
  __shared__ __align__(16) float B1[NR * PSTR];  // Kcur / P2 / P10 / Mplus
  __shared__ __align__(16) float B2[NR * PSTR];  // Ssum / P4 / P20
  __shared__ __align__(16) float B3[NR * PSTR];  // tmp  / Sh
  __shared__ __align__(16) float WpT[DD * WSTR]; // W_pen^T  [84][128]

  const int tid  = threadIdx.x;
  const int NT   = blockDim.x;
  const int wave = tid >> 5;
  const int lane = tid & 31;
  const int m    = lane & 15;
  const int koff = (lane < 16) ? 0 : 2;
  const int half = (lane < 16) ? 0 : 8;
  const float h  = HSTEP;
  const v8f zero8 = {0.f, 0.f, 0.f, 0.f, 0.f, 0.f, 0.f, 0.f};

  for (int i = tid; i < NR * PSTR; i += NT) { B0[i] = 0.f; B1[i] = 0.f; B2[i] = 0.f; B3[i] = 0.f; }
  __syncthreads();

  // Aaug: A[r][c] = W[c][r] (row-vector convention: f(z) = z @ A + b); row 84 = b
  for (int i = tid; i < DD * DD; i += NT) {
    int r = i / DD, c = i % DD;
    B0[r * PSTR + c] = Wode[c * DD + r];
  }
  for (int j = tid; j < DD; j += NT) B0[DD * PSTR + j] = bode[j];
  // Stage W_pen^T: WpT[q][j] = Wpen[j][q]
  for (int i = tid; i < DD * HH; i += NT) {
    int q = i / HH, j = i % HH;
    WpT[q * WSTR + j] = Wpen[j * DD + q];
  }
  __syncthreads();

  // K1 = Aaug; Ssum = K1
  for (int i = tid; i < NR * PSTR; i += NT) { B1[i] = B0[i]; B2[i] = B0[i]; }
  __syncthreads();

  // WMMA matmul over the 96x96 padded square: Z = X @ Y, 36 tiles over 8 waves.
  // (tile loop condition is wave-uniform -> EXEC all-ones at every WMMA)
#define MM_WMMA(X, Y, Z, KSTEPS)                                               \
  for (int tile = wave; tile < 36; tile += 8) {                                \
    const int tr = tile / 6, tc = tile % 6;                                    \
    v8f acc = zero8;                                                           \
    for (int ks = 0; ks < (KSTEPS); ++ks) {                                    \
      const int k = ks * 4;                                                    \
      v2f a, b;                                                                \
      a.x = (X)[(tr * 16 + m) * PSTR + k + koff];                              \
      a.y = (X)[(tr * 16 + m) * PSTR + k + koff + 1];                          \
      b.x = (Y)[(k + koff) * PSTR + tc * 16 + m];                              \
      b.y = (Y)[(k + koff + 1) * PSTR + tc * 16 + m];                          \
      acc = __builtin_amdgcn_wmma_f32_16x16x4_f32(                             \
          false, a, false, b, (short)0, acc, false, false);                    \
    }                                                                          \
    _Pragma("unroll")                                                          \
    for (int r = 0; r < 8; ++r)                                                \
      (Z)[(tr * 16 + r + half) * PSTR + tc * 16 + m] = acc[r];                 \
  }                                                                            \
  __syncthreads();

  // K_{i+1} = Aaug + coef * (K_i @ A);  Ssum += wgt * K_{i+1}   (K = 84 exact)
  const float coef[3] = {0.5f * h, 0.5f * h, h};
  const float wgt[3]  = {2.f, 2.f, 1.f};
  for (int s = 0; s < 3; ++s) {
    MM_WMMA(B1, B0, B3, 21)                       // tmp = Kcur @ A
    for (int i = tid; i < DH * DD; i += NT) {
      int r = i / DD, c = i % DD;
      float kn = B0[r * PSTR + c] + coef[s] * B3[r * PSTR + c];
      B1[r * PSTR + c] = kn;
      B2[r * PSTR + c] += wgt[s] * kn;
    }
    __syncthreads();
  }

  // Homogeneous per-step map Sh (85x85) into B3:
  // rows<84: I + (h/6)Ssum ; row84: [(h/6)Ssum_bias , 1]
  for (int i = tid; i < DH * DH; i += NT) {
    int r = i / DH, c = i % DH;
    float v;
    if (c < DD) v = ((r == c) ? 1.f : 0.f) + (h / 6.f) * B2[r * PSTR + c];
    else        v = (r == DD) ? 1.f : 0.f;
    B3[r * PSTR + c] = v;
  }
  __syncthreads();

  // Sh^20: S2=Sh*Sh, S4=S2*S2, S5=S4*Sh, S10=S5*S5, S20=S10*S10   (K = 88)
  MM_WMMA(B3, B3, B1, 22)   // P2
  MM_WMMA(B1, B1, B2, 22)   // P4
  MM_WMMA(B2, B3, B0, 22)   // P5
  MM_WMMA(B0, B0, B1, 22)   // P10
  MM_WMMA(B1, B1, B2, 22)   // P20 = [[T,0],[c,1]]

  // Mplus = P20 + I (diag only on state rows; bias row 84 untouched)
  for (int i = tid; i < NR * PSTR; i += NT) B1[i] = B2[i];
  __syncthreads();
  for (int r = tid; r < DD; r += NT) B1[r * PSTR + r] += 1.f;
  __syncthreads();

  // E = Mplus @ W_pen^T : 6x8 tiles, K = 84.  Rows 0..75 -> g_E, row 84 -> g_bp.
  for (int tile = wave; tile < 48; tile += 8) {
    const int tr = tile / 8, tc = tile % 8;
    v8f acc = zero8;
    for (int ks = 0; ks < 21; ++ks) {
      const int k = ks * 4;
      v2f a, b;
      a.x = B1[(tr * 16 + m) * PSTR + k + koff];
      a.y = B1[(tr * 16 + m) * PSTR + k + koff + 1];
      b.x = WpT[(k + koff) * WSTR + tc * 16 + m];
      b.y = WpT[(k + koff + 1) * WSTR + tc * 16 + m];
      acc = __builtin_amdgcn_wmma_f32_16x16x4_f32(
          false, a, false, b, (short)0, acc, false, false);
    }
#pragma unroll
    for (int r = 0; r < 8; ++r) {
      const int row = tr * 16 + r + half;
      const int col = tc * 16 + m;
      if (row < FF)       g_E[row * HH + col] = acc[r];
      else if (row == DD) g_bp[col] = acc[r] + bpen[col];
    }
  }
}

// ---------------------------------------------------------------------------
// Kernel 2: out = relu(x @ E + b') @ Wfin^T + bfin via V_WMMA_F32_16X16X4_F32
// 128 rows/block, 8 waves, each wave owns one 16-row tile x full 128 cols.
// ---------------------------------------------------------------------------
#define RPB   128   // rows per block
#define ESTR  132   // E LDS stride (132 % 64 = 4 -> half-wave K rows hit distinct banks)
#define HSTR  132   // h LDS stride (lane-m stride 132 -> banks 4m, conflict-free)

__global__ __launch_bounds__(256) void anode_main(
    const float* __restrict__ x, const float* __restrict__ Wfin,
    const float* __restrict__ bfin, float* __restrict__ out, int Nrows)
{
  __shared__ __align__(16) float xs[RPB * FF];        // 38,912 B
  __shared__ __align__(16) float Es[FF * ESTR];       // 40,128 B
  __shared__ __align__(16) float wfs[HH * 16];        //  8,192 B  (Wfin^T, cols 5..15 zero)
  __shared__ __align__(16) float hs[8 * 16 * HSTR];   // 67,584 B  (per-wave h tiles)

  const int tid = threadIdx.x;
  const int r0  = blockIdx.x * RPB;

  // Stage x tile (rows guarded, float4 = global_load_b128 + ds_store_b128)
  for (int i = tid; i < RPB * (FF / 4); i += 256) {
    int row = i / (FF / 4), q = i % (FF / 4);
    float4 v = make_float4(0.f, 0.f, 0.f, 0.f);
    if (r0 + row < Nrows) v = ((const float4*)x)[(size_t)(r0 + row) * (FF / 4) + q];
    *(float4*)&xs[row * FF + q * 4] = v;
  }
  // Stage E (L2-hot, 38.9 KB)
  for (int i = tid; i < FF * (HH / 4); i += 256) {
    int rr = i / (HH / 4), q = i % (HH / 4);
    float4 v = ((const float4*)g_E)[rr * (HH / 4) + q];
    *(float4*)&Es[rr * ESTR + q * 4] = v;
  }
  // Stage Wfin^T into zero-padded 16-wide tile
  for (int i = tid; i < HH * 16; i += 256) {
    int k = i >> 4, n = i & 15;
    wfs[i] = (n < CC) ? Wfin[n * HH + k] : 0.f;
  }
  __syncthreads();

  const int wave = tid >> 5;
  const int lane = tid & 31;
  const int m    = lane & 15;                 // A: M index / B,C,D: N index
  const int koff = (lane < 16) ? 0 : 2;       // VGPR-half selects K pair
  const int rowb = wave * 16;

  const v8f zero8 = {0.f, 0.f, 0.f, 0.f, 0.f, 0.f, 0.f, 0.f};
  v8f acc[8];
#pragma unroll
  for (int t = 0; t < 8; ++t) acc[t] = zero8;

  // GEMM1: (16x76) @ (76x128), K = 19 steps of 4, A-frag reused across 8 N-tiles
  for (int ks = 0; ks < 19; ++ks) {
    const int k = ks * 4;
    v2f a;
    a.x = xs[(rowb + m) * FF + k + koff];
    a.y = xs[(rowb + m) * FF + k + koff + 1];
#pragma unroll
    for (int t = 0; t < 8; ++t) {
      v2f b;
      b.x = Es[(k + koff)     * ESTR + t * 16 + m];
      b.y = Es[(k + koff + 1) * ESTR + t * 16 + m];
      acc[t] = __builtin_amdgcn_wmma_f32_16x16x4_f32(
          false, a, false, b, (short)0, acc[t], false, false);
    }
  }

  // Fused bias + relu, re-layout C-frags -> A-frags via per-wave LDS tile
#pragma unroll
  for (int t = 0; t < 8; ++t) {
    float bp = g_bp[t * 16 + m];
#pragma unroll
    for (int r = 0; r < 8; ++r) {
      float v = acc[t][r] + bp;
      v = v > 0.f ? v : 0.f;
      int row = r + ((lane < 16) ? 0 : 8);
      hs[(wave * 16 + row) * HSTR + t * 16 + m] = v;
    }
  }
  __syncthreads();

  // GEMM2: (16x128) @ (128x16[5 valid]), K = 32 steps of 4
  v8f acc2 = zero8;
  for (int ks = 0; ks < 32; ++ks) {
    const int k = ks * 4;
    v2f a;
    a.x = hs[(wave * 16 + m) * HSTR + k + koff];
    a.y = hs[(wave * 16 + m) * HSTR + k + koff + 1];
    v2f b;
    b.x = wfs[(k + koff)     * 16 + m];
    b.y = wfs[(k + koff + 1) * 16 + m];
    acc2 = __builtin_amdgcn_wmma_f32_16x16x4_f32(
        false, a, false, b, (short)0, acc2, false, false);
  }

  // Store the 5 valid output columns
  if (m < CC) {
    float bf = bfin[m];
#pragma unroll
    for (int r = 0; r < 8; ++r) {
      int row = r0 + rowb + r + ((lane < 16) ? 0 : 8);
      if (row < Nrows) out[(size_t)row * CC + m] = acc2[r] + bf;
    }
  }
}

// ---------------------------------------------------------------------------
extern "C" void kernel_launch(void* const* d_in, const int* in_sizes, int n_in,
                              void* d_out, int out_size, void* d_ws, size_t ws_size,
                              hipStream_t stream) {
  (void)n_in; (void)d_ws; (void)ws_size; (void)out_size;
  const float* x    = (const float*)d_in[0];
  const float* Wode = (const float*)d_in[1];
  const float* bode = (const float*)d_in[2];
  const float* Wpen = (const float*)d_in[3];
  const float* bpen = (const float*)d_in[4];
  const float* Wfin = (const float*)d_in[5];
  const float* bfin = (const float*)d_in[6];
  float* out = (float*)d_out;

  const int Nrows = in_sizes[0] / FF;  // 32*2048 = 65536

  anode_precompute<<<1, 256, 0, stream>>>(Wode, bode, Wpen, bpen);
  const int nblocks = (Nrows + RPB - 1) / RPB;
  anode_main<<<nblocks, 256, 0, stream>>>(x, Wfin, bfin, out, Nrows);
}